// DiffMambaBackbone_1262720385716
// MI455X (gfx1250) — compile-verified
//
#include <hip/hip_runtime.h>
#include <math.h>

typedef float v2f __attribute__((ext_vector_type(2)));
typedef float v8f __attribute__((ext_vector_type(8)));

#define DEV static __device__ __forceinline__

DEV float silu_(float x) { return x / (1.0f + __expf(-x)); }

DEV float wave_sum32(float v) {
  v += __shfl_xor(v, 16, 32);
  v += __shfl_xor(v, 8, 32);
  v += __shfl_xor(v, 4, 32);
  v += __shfl_xor(v, 2, 32);
  v += __shfl_xor(v, 1, 32);
  return v;
}

DEV v8f wmma4(v2f a, v2f b, v8f c) {
  return __builtin_amdgcn_wmma_f32_16x16x4_f32(false, a, false, b,
                                               (short)0, c, false, false);
}

// ---------------------------------------------------------------------------
// WMMA f32 GEMM, 2x2 register-blocked: each wave32 computes a 32x32 C tile
// (4 x v8f accumulators) via v_wmma_f32_16x16x4_f32. Per K-step: 2 A frags +
// 2 B frags -> 4 WMMAs (2x arithmetic intensity vs 1 tile/wave).
//   A frag (16x4 f32): lane m=lane&15, v0/v1 hold K = 2*(lane>>4) + {0,1}
//   B frag (4x16 f32): VGPR j, lanes0-15 -> K=j, lanes16-31 -> K=j+2
//   C frag: VGPR r, lanes0-15 -> M=r, lanes16-31 -> M=r+8, N=lane&15
// K % 4 == 0, N % 16 == 0 guaranteed by the model shapes; M and the ragged
// N=48 group are edge-masked with clamped pointers (EXEC stays all-1s).
// Epilogues: 0=store, 1=add-into-C (residual), 2=bias+softplus
// ---------------------------------------------------------------------------
template <int EP>
__global__ void k_gemm(const float* __restrict__ A, int lda,
                       const float* __restrict__ B, int ldb,
                       float* __restrict__ C, int ldc,
                       int M, int N, int K, const float* __restrict__ bias) {
  int wave = (blockIdx.x * blockDim.x + threadIdx.x) >> 5;
  int lane = threadIdx.x & 31;
  int nt = (N + 31) >> 5;
  int tm = wave / nt;
  int tn = wave - tm * nt;
  if (tm * 32 >= M) return;  // wave-uniform

  int m0 = tm * 32 + (lane & 15), m1 = m0 + 16;
  int n0 = tn * 32 + (lane & 15), n1 = n0 + 16;
  int khi = (lane >> 4) << 1;  // 0 or 2
  bool mv0 = (m0 < M), mv1 = (m1 < M);
  bool nv1 = (n1 < N);  // n0 always valid (N % 16 == 0)
  float ms0 = mv0 ? 1.0f : 0.0f, ms1 = mv1 ? 1.0f : 0.0f;
  const float* Ar0 = A + (size_t)(mv0 ? m0 : 0) * lda;
  const float* Ar1 = A + (size_t)(mv1 ? m1 : 0) * lda;
  int nc1 = nv1 ? n1 : n0;  // clamped load col; store predicated on nv1

  v8f a00 = {}, a01 = {}, a10 = {}, a11 = {};
  for (int kb = 0; kb < K; kb += 4) {
    int k0 = kb + khi;
    const float* Br0 = B + (size_t)k0 * ldb;
    const float* Br1 = B + (size_t)(k0 + 1) * ldb;
    v2f af0, af1, bf0, bf1;
    af0.x = Ar0[k0] * ms0;  af0.y = Ar0[k0 + 1] * ms0;
    af1.x = Ar1[k0] * ms1;  af1.y = Ar1[k0 + 1] * ms1;
    bf0.x = Br0[n0];        bf0.y = Br1[n0];
    bf1.x = Br0[nc1];       bf1.y = Br1[nc1];
    a00 = wmma4(af0, bf0, a00);
    a01 = wmma4(af0, bf1, a01);
    a10 = wmma4(af1, bf0, a10);
    a11 = wmma4(af1, bf1, a11);
  }

  int rb0 = tm * 32 + ((lane >> 4) << 3);
  int rb1 = rb0 + 16;
  int c0 = n0, c1 = n1;

  auto store_tile = [&](const v8f& acc, int rbase, int col, bool colv) {
#pragma unroll
    for (int r = 0; r < 8; r++) {
      int row = rbase + r;
      if (colv && row < M) {
        size_t off = (size_t)row * ldc + col;
        float v = acc[r];
        if (EP == 1) {
          C[off] += v;
        } else if (EP == 2) {
          v += bias[col];
          C[off] = (v > 20.0f) ? v : log1pf(__expf(v));
        } else {
          C[off] = v;
        }
      }
    }
  };
  store_tile(a00, rb0, c0, true);
  store_tile(a01, rb0, c1, nv1);
  store_tile(a10, rb1, c0, true);
  store_tile(a11, rb1, c1, nv1);
}

// ---------------------------------------------------------------------------
// Direct conv2d patch embed. Output layout [B, L=Hout*Wout, Cout].
// Optional second conv (stage 0 mask path) added in.
// ---------------------------------------------------------------------------
__global__ void k_conv2d(const float* __restrict__ in, const float* __restrict__ w,
                         const float* __restrict__ bias,
                         const float* __restrict__ in2, const float* __restrict__ w2,
                         const float* __restrict__ bias2,
                         float* __restrict__ out,
                         int Bn, int Cin, int Hin, int Win,
                         int Cout, int Hout, int Wout, int k, int s, int p) {
  int idx = blockIdx.x * blockDim.x + threadIdx.x;
  int total = Bn * Cout * Hout * Wout;
  if (idx >= total) return;
  int co = idx % Cout; int t = idx / Cout;
  int xw = t % Wout; t /= Wout;
  int xh = t % Hout; int b = t / Hout;
  float acc = bias[co];
  for (int ci = 0; ci < Cin; ci++)
    for (int kh = 0; kh < k; kh++) {
      int hi = xh * s - p + kh;
      if (hi < 0 || hi >= Hin) continue;
      for (int kw = 0; kw < k; kw++) {
        int wi = xw * s - p + kw;
        if (wi < 0 || wi >= Win) continue;
        acc += in[(((size_t)b * Cin + ci) * Hin + hi) * Win + wi] *
               w[((co * Cin + ci) * k + kh) * k + kw];
      }
    }
  if (in2) {
    acc += bias2[co];
    for (int kh = 0; kh < k; kh++) {
      int hi = xh * s - p + kh;
      if (hi < 0 || hi >= Hin) continue;
      for (int kw = 0; kw < k; kw++) {
        int wi = xw * s - p + kw;
        if (wi < 0 || wi >= Win) continue;
        acc += in2[((size_t)b * Hin + hi) * Win + wi] * w2[(co * k + kh) * k + kw];
      }
    }
  }
  out[((size_t)b * Hout * Wout + (xh * Wout + xw)) * Cout + co] = acc;
}

// ---------------------------------------------------------------------------
// LayerNorm over last dim, one wave32 per row. Optional AdaLN modulation
// (h*(1+te)+te) and optional NCHW-transposed store.
// ---------------------------------------------------------------------------
__global__ void k_layernorm(const float* __restrict__ x, const float* __restrict__ g,
                            const float* __restrict__ bb, const float* __restrict__ te,
                            float* __restrict__ out, int rows, int L, int C, int ochw) {
  int wave = (blockIdx.x * blockDim.x + threadIdx.x) >> 5;
  int lane = threadIdx.x & 31;
  if (wave >= rows) return;
  const float* xr = x + (size_t)wave * C;
  float s = 0.f;
  for (int c = lane; c < C; c += 32) s += xr[c];
  float mu = wave_sum32(s) / (float)C;
  float v = 0.f;
  for (int c = lane; c < C; c += 32) { float d = xr[c] - mu; v += d * d; }
  float inv = 1.0f / sqrtf(wave_sum32(v) / (float)C + 1e-5f);
  int b = wave / L, l = wave - b * L;
  for (int c = lane; c < C; c += 32) {
    float y = (xr[c] - mu) * inv * g[c] + bb[c];
    if (te) { float t = te[b * C + c]; y = y * (1.0f + t) + t; }
    if (ochw) out[((size_t)b * C + c) * L + l] = y;
    else      out[(size_t)wave * C + c] = y;
  }
}

// ---------------------------------------------------------------------------
// Timestep embedding MLP (B=2 rows): freq(256) -> d (silu) -> d
// ---------------------------------------------------------------------------
__global__ void k_timeembed(const float* __restrict__ tsteps,
                            const float* __restrict__ w1, const float* __restrict__ b1,
                            const float* __restrict__ w2, const float* __restrict__ b2,
                            float* __restrict__ te, int d) {
  extern __shared__ float sm[];  // 256 + d
  float* emb = sm;
  float* h1  = sm + 256;
  int b = blockIdx.x, tid = threadIdx.x;
  float t = tsteps[b];
  for (int j = tid; j < 128; j += blockDim.x) {
    float fr = __expf(-9.210340371976184f * (float)j / 128.0f);  // -ln(1e4)
    float a = t * fr;
    emb[j] = cosf(a);
    emb[128 + j] = sinf(a);
  }
  __syncthreads();
  for (int j = tid; j < d; j += blockDim.x) {
    float h = b1[j];
    for (int kk = 0; kk < 256; kk++) h += emb[kk] * w1[kk * d + j];
    h1[j] = silu_(h);
  }
  __syncthreads();
  for (int j = tid; j < d; j += blockDim.x) {
    float v = b2[j];
    for (int kk = 0; kk < d; kk++) v += h1[kk] * w2[kk * d + j];
    te[b * d + j] = v;
  }
}

// ---------------------------------------------------------------------------
// Causal depthwise conv1d (K=4) + bias + SiLU. u cols live at [0,E) of hbuf
// rows of stride 2E. Output [B,L,E].
// ---------------------------------------------------------------------------
__global__ void k_dwconv_silu(const float* __restrict__ h, const float* __restrict__ cw,
                              const float* __restrict__ cb, float* __restrict__ u,
                              int Bn, int L, int E) {
  int idx = blockIdx.x * blockDim.x + threadIdx.x;
  int total = Bn * L * E;
  if (idx >= total) return;
  int e = idx % E; int t = idx / E;
  int l = t % L;   int b = t / L;
  int S = 2 * E;
  float acc = cb[e];
#pragma unroll
  for (int j = 0; j < 4; j++) {
    int ls = l - 3 + j;
    if (ls >= 0) acc += cw[e * 4 + j] * h[((size_t)b * L + ls) * S + e];
  }
  u[((size_t)b * L + l) * E + e] = silu_(acc);
}

// ---------------------------------------------------------------------------
// Selective scan: each 16-lane half-wave owns one (b,e); lanes carry the
// N=16 state; cross-lane xor-reduce produces y each step. y += u*D fused.
// ---------------------------------------------------------------------------
__global__ void k_scan(const float* __restrict__ dt, const float* __restrict__ u,
                       const float* __restrict__ dtbc, const float* __restrict__ Alog,
                       const float* __restrict__ Dp, float* __restrict__ y,
                       int Bn, int L, int E) {
  int wave = (blockIdx.x * blockDim.x + threadIdx.x) >> 5;
  int lane = threadIdx.x & 31;
  int half = lane >> 4, n = lane & 15;
  int pair = wave * 2 + half;
  if (pair >= Bn * E) return;  // Bn*E even -> wave-uniform
  int b = pair / E, e = pair - b * E;
  float A  = -__expf(Alog[e * 16 + n]);
  float De = Dp[e];
  float h = 0.0f;
  const float* dtp = dt + (size_t)b * L * E + e;
  const float* up  = u  + (size_t)b * L * E + e;
  const float* bc  = dtbc + (size_t)b * L * 48 + 16 + n;
  float* yp = y + (size_t)b * L * E + e;
  for (int l = 0; l < L; l++) {
    float dtl = dtp[(size_t)l * E];
    float ul  = up[(size_t)l * E];
    float Bv  = bc[(size_t)l * 48];
    float Cv  = bc[(size_t)l * 48 + 16];
    h = __expf(dtl * A) * h + dtl * Bv * ul;
    float pv = h * Cv;
    pv += __shfl_xor(pv, 1, 32);
    pv += __shfl_xor(pv, 2, 32);
    pv += __shfl_xor(pv, 4, 32);
    pv += __shfl_xor(pv, 8, 32);
    if (n == 0) yp[(size_t)l * E] = pv + ul * De;
  }
}

// y *= silu(z); z at col offset E of hbuf rows (stride 2E)
__global__ void k_gate(float* __restrict__ y, const float* __restrict__ h,
                       int Bn, int L, int E) {
  int idx = blockIdx.x * blockDim.x + threadIdx.x;
  int total = Bn * L * E;
  if (idx >= total) return;
  int e = idx % E; int t = idx / E;
  int l = t % L;   int b = t / L;
  float z = h[((size_t)b * L + l) * (size_t)(2 * E) + E + e];
  y[idx] = y[idx] * silu_(z);
}

// ===========================================================================
// Host orchestration
// ===========================================================================
struct BlkP {
  const float *Alog, *D, *conv_b, *conv_w, *dt_b, *dt_w, *in_w, *ln_b, *ln_g,
      *out_w, *xproj_w;
};
struct StgP {
  const float *mask_b, *mask_w, *norm_b, *norm_g, *pe_b, *pe_g, *proj_b,
      *proj_w, *t_b1, *t_b2, *t_w1, *t_w2;
  BlkP blk[4];
};

extern "C" void kernel_launch(void* const* d_in, const int* in_sizes, int n_in,
                              void* d_out, int out_size, void* d_ws, size_t ws_size,
                              hipStream_t stream) {
  (void)in_sizes; (void)n_in; (void)out_size; (void)ws_size;
  const float* x      = (const float*)d_in[0];
  const float* tsteps = (const float*)d_in[1];
  const float* cond   = (const float*)d_in[2];

  // params flattened as JAX tree leaves (dict keys sorted)
  static const int DIMS[4]   = {64, 128, 320, 512};
  static const int DEPTH[4]  = {2, 2, 4, 2};
  static const int KS[4]     = {4, 2, 2, 2};
  static const int STRIDE[4] = {4, 2, 2, 2};
  static const int PADS[4]   = {2, 1, 1, 1};

  int pi = 3;
  auto nxt = [&]() { return (const float*)d_in[pi++]; };
  StgP st[4];
  for (int i = 0; i < 4; i++) {
    for (int j = 0; j < DEPTH[i]; j++) {
      BlkP& bp = st[i].blk[j];
      bp.Alog = nxt(); bp.D = nxt(); bp.conv_b = nxt(); bp.conv_w = nxt();
      bp.dt_b = nxt(); bp.dt_w = nxt(); bp.in_w = nxt(); bp.ln_b = nxt();
      bp.ln_g = nxt(); bp.out_w = nxt(); bp.xproj_w = nxt();
    }
    if (i == 0) { st[i].mask_b = nxt(); st[i].mask_w = nxt(); }
    else        { st[i].mask_b = nullptr; st[i].mask_w = nullptr; }
    st[i].norm_b = nxt(); st[i].norm_g = nxt();
    st[i].pe_b = nxt();   st[i].pe_g = nxt();
    st[i].proj_b = nxt(); st[i].proj_w = nxt();
    st[i].t_b1 = nxt(); st[i].t_b2 = nxt(); st[i].t_w1 = nxt(); st[i].t_w2 = nxt();
  }

  // workspace partition (floats)
  float* W = (float*)d_ws;
  float* seq   = W;             // <= 1,100,000
  float* hmod  = W + 1100000;   // <= 1,100,000 (also conv staging)
  float* hbuf  = W + 2200000;   // <= 4,100,000 ([M,4d]: u|z)
  float* ubuf  = W + 6300000;   // <= 2,100,000
  float* dtbcb = W + 8400000;   // <=   800,000 ([M,48]: dt|B|C)
  float* dtb   = W + 9200000;   // <= 2,100,000
  float* ybuf  = W + 11300000;  // <= 2,100,000
  float* teb   = W + 13400000;  // <= 1,024

  float* outp = (float*)d_out;
  const size_t outoff[4] = {0, 1013888, 1532288, 1870848};

  auto gemm = [&](int ep, const float* A, int lda, const float* B, int ldb,
                  float* C, int ldc, int M, int N, int K, const float* bias) {
    int mt = (M + 31) / 32, nt = (N + 31) / 32;
    long waves = (long)mt * nt;
    int blocks = (int)((waves + 7) / 8);
    dim3 blk(256);
    if (ep == 0)      hipLaunchKernelGGL((k_gemm<0>), dim3(blocks), blk, 0, stream, A, lda, B, ldb, C, ldc, M, N, K, bias);
    else if (ep == 1) hipLaunchKernelGGL((k_gemm<1>), dim3(blocks), blk, 0, stream, A, lda, B, ldb, C, ldc, M, N, K, bias);
    else              hipLaunchKernelGGL((k_gemm<2>), dim3(blocks), blk, 0, stream, A, lda, B, ldb, C, ldc, M, N, K, bias);
  };

  int Hin = 352, Win = 352, curC = 3;
  const float* curNCHW = cond;

  for (int i = 0; i < 4; i++) {
    const StgP& sp = st[i];
    int d = DIMS[i], E = 2 * d, S = 4 * d;
    int k = KS[i], s = STRIDE[i], p = PADS[i];
    int Hout = (Hin + 2 * p - k) / s + 1, Wout = Hout;
    int L = Hout * Wout, M = 2 * L;

    // patch embed conv -> [B,L,d] in hmod (stage 0 adds mask conv of x)
    {
      int total = 2 * d * Hout * Wout;
      k_conv2d<<<(total + 255) / 256, 256, 0, stream>>>(
          curNCHW, sp.proj_w, sp.proj_b,
          (i == 0) ? x : nullptr, sp.mask_w, sp.mask_b,
          hmod, 2, curC, Hin, Win, d, Hout, Wout, k, s, p);
    }
    // pos-embed layernorm -> seq
    k_layernorm<<<(M + 3) / 4, 128, 0, stream>>>(hmod, sp.pe_g, sp.pe_b, nullptr,
                                                 seq, M, L, d, 0);
    // timestep embedding -> teb [2,d]
    k_timeembed<<<2, d, (256 + d) * sizeof(float), stream>>>(
        tsteps, sp.t_w1, sp.t_b1, sp.t_w2, sp.t_b2, teb, d);

    for (int j = 0; j < DEPTH[i]; j++) {
      const BlkP& bp = sp.blk[j];
      // LN + AdaLN modulation -> hmod
      k_layernorm<<<(M + 3) / 4, 128, 0, stream>>>(seq, bp.ln_g, bp.ln_b, teb,
                                                   hmod, M, L, d, 0);
      // in_proj: [M,d]@[d,4d] -> hbuf (u|z)
      gemm(0, hmod, d, bp.in_w, S, hbuf, S, M, S, d, nullptr);
      // causal depthwise conv + silu -> ubuf
      {
        int total = M * E;
        k_dwconv_silu<<<(total + 255) / 256, 256, 0, stream>>>(hbuf, bp.conv_w,
                                                               bp.conv_b, ubuf, 2, L, E);
      }
      // x_proj: [M,E]@[E,48] -> dtbcb (dt|B|C)
      gemm(0, ubuf, E, bp.xproj_w, 48, dtbcb, 48, M, 48, E, nullptr);
      // dt: softplus([M,16]@[16,E] + dt_b) -> dtb
      gemm(2, dtbcb, 48, bp.dt_w, E, dtb, E, M, E, 16, bp.dt_b);
      // selective scan (+ u*D) -> ybuf
      k_scan<<<(E + 3) / 4, 128, 0, stream>>>(dtb, ubuf, dtbcb, bp.Alog, bp.D,
                                              ybuf, 2, L, E);
      // gate: y *= silu(z)
      {
        int total = M * E;
        k_gate<<<(total + 255) / 256, 256, 0, stream>>>(ybuf, hbuf, 2, L, E);
      }
      // out_proj + residual: seq += [M,E]@[E,d]
      gemm(1, ybuf, E, bp.out_w, d, seq, d, M, d, E, nullptr);
    }

    // final norm, written NCHW straight into d_out (also next stage input)
    k_layernorm<<<(M + 3) / 4, 128, 0, stream>>>(seq, sp.norm_g, sp.norm_b, nullptr,
                                                 outp + outoff[i], M, L, d, 1);
    curNCHW = outp + outoff[i];
    curC = d; Hin = Hout; Win = Wout;
  }
}